// TtMambaSSM_17162689315583
// MI455X (gfx1250) — compile-verified
//
#include <hip/hip_runtime.h>
#include <hip/hip_bf16.h>

// ---- problem constants (from reference) ----
#define U_DIM 32
#define H_DIM 5120
#define R_DIM 160
#define N_DIM 16

typedef float v2f __attribute__((ext_vector_type(2)));
typedef float v4f __attribute__((ext_vector_type(4)));
typedef float v8f __attribute__((ext_vector_type(8)));

// Workspace layout (floats)
//  part : [0, 61440)               12 tiles x 10 ksplits x 512
//  dlow : [61440, 66560)           32 x 160
//  Bf   : [66560, 67072)           32 x 16
//  Cf   : [67072, 67584)           32 x 16
#define WS_PART  0
#define WS_DLOW  61440
#define WS_BF    66560
#define WS_CF    67072

// ---------------------------------------------------------------------------
// Kernel 1: partial split-K GEMM   out-tiles: t=0..9 -> x@W_delta (16 cols),
// t=10 -> x@W_B, t=11 -> x@W_C.  Grid = 12 tiles * 10 ksplits = 120 blocks.
// Per block: 8 waves; wave w: M-tile m=w&1 (rows u=m*16..+15),
// K sub-chunk kw=w>>1 of 128 (32 WMMA_F32_16X16X4 steps).
// ---------------------------------------------------------------------------
__global__ void k1_partial(const float* __restrict__ x,
                           const float* __restrict__ W_delta,
                           const float* __restrict__ W_B,
                           const float* __restrict__ W_C,
                           float* __restrict__ part)
{
    const int t    = blockIdx.x % 12;
    const int s    = blockIdx.x / 12;
    const int tid  = threadIdx.x;
    const int lane = tid & 31;
    const int wave = tid >> 5;      // 0..7
    const int m    = wave & 1;      // M tile
    const int kw   = wave >> 1;     // 0..3

    const int lhalf = (lane < 16) ? 0 : 2;  // K sub-offset per lane half
    const int lr    = lane & 15;

    // uniform (scalar) selection of the B operand matrix
    const float* W;
    int ldw, c0;
    if (t < 10)       { W = W_delta; ldw = R_DIM; c0 = t * 16; }
    else if (t == 10) { W = W_B;     ldw = N_DIM; c0 = 0;      }
    else              { W = W_C;     ldw = N_DIM; c0 = 0;      }

    const int row       = m * 16 + lr;            // x row (M index) this lane feeds
    const float* xr     = x + (size_t)row * H_DIM;

    v8f acc = {0.f,0.f,0.f,0.f,0.f,0.f,0.f,0.f};
    int kb = s * 512 + kw * 128;
    #pragma unroll 4
    for (int it = 0; it < 32; ++it, kb += 4) {
        v2f a, b;
        a.x = xr[kb + lhalf];
        a.y = xr[kb + lhalf + 1];
        b.x = W[(size_t)(kb + lhalf    ) * ldw + c0 + lr];
        b.y = W[(size_t)(kb + lhalf + 1) * ldw + c0 + lr];
        acc = __builtin_amdgcn_wmma_f32_16x16x4_f32(
                  false, a, false, b, (short)0, acc, false, false);
    }

    // staged (deterministic) reduction of the 4 K-chunk partials per M tile
    __shared__ float lds[8 * 256];
    #pragma unroll
    for (int i = 0; i < 8; ++i) {
        const int r = i + ((lane < 16) ? 0 : 8);          // C-matrix row layout
        lds[wave * 256 + r * 16 + lr] = acc[i];
    }
    __syncthreads();

    #pragma unroll
    for (int j = 0; j < 2; ++j) {
        const int p   = tid + j * 256;  // 0..511
        const int m2  = p >> 8;
        const int idx = p & 255;
        const float ssum = lds[(0 + m2) * 256 + idx]
                         + lds[(2 + m2) * 256 + idx]
                         + lds[(4 + m2) * 256 + idx]
                         + lds[(6 + m2) * 256 + idx];
        part[(size_t)blockIdx.x * 512 + p] = ssum;
    }
}

// ---------------------------------------------------------------------------
// Kernel 1b: deterministic reduction over the 10 K-splits.
// 6144 outputs -> 24 blocks x 256 threads.
// ---------------------------------------------------------------------------
__global__ void k1_reduce(const float* __restrict__ part,
                          float* __restrict__ dlow,
                          float* __restrict__ Bf,
                          float* __restrict__ Cf)
{
    const int o = blockIdx.x * 256 + threadIdx.x;   // 0..6143
    const int t = o / 512;
    const int r = o % 512;
    float s = 0.f;
    #pragma unroll
    for (int ks = 0; ks < 10; ++ks)
        s += part[(size_t)(ks * 12 + t) * 512 + r];
    const int m   = r >> 8;
    const int idx = r & 255;
    const int row = idx >> 4;
    const int col = idx & 15;
    const int u   = m * 16 + row;
    if (t < 10)       dlow[u * R_DIM + t * 16 + col] = s;
    else if (t == 10) Bf[u * N_DIM + col] = s;
    else              Cf[u * N_DIM + col] = s;
}

// ---------------------------------------------------------------------------
// Kernel 2: fused  dt = softplus(dlow@W_dt + b_dt)  +  SSM state update + mix.
// One block per 16 H-channels (320 blocks, 256 threads = 8 waves).
// Phase A: WMMA f32 16x16x4, K=160 split over 4 wave-pairs, LDS staged reduce.
// Phase B: each thread handles 2 (u, h-channel) pairs, reads its 64B h slice
//          once (non-temporal v4f), does exp/softplus math, writes y.
// h loads are prefetched before the WMMA phase (no data dependence on dt).
// ---------------------------------------------------------------------------
__global__ void k2_fused(const float* __restrict__ x,
                         const float* __restrict__ W_dt,
                         const float* __restrict__ b_dt,
                         const float* __restrict__ A_log,
                         const float* __restrict__ D,
                         const float* __restrict__ h,
                         const float* __restrict__ dlow,
                         const float* __restrict__ Bf,
                         const float* __restrict__ Cf,
                         float* __restrict__ yout)
{
    const int h0   = blockIdx.x * 16;   // H tile
    const int tid  = threadIdx.x;
    const int lane = tid & 31;
    const int wave = tid >> 5;
    const int m    = wave & 1;
    const int kw   = wave >> 1;         // 0..3, K chunk of 40

    // prefetch this block's h tile (independent of dt) to overlap with WMMA
    #pragma unroll
    for (int j = 0; j < 2; ++j) {
        const int p  = tid + j * 256;
        const int hh = p & 15;
        const int u  = p >> 4;
        __builtin_prefetch(h + ((size_t)u * H_DIM + h0 + hh) * N_DIM, 0, 0);
    }

    const int lhalf = (lane < 16) ? 0 : 2;
    const int lr    = lane & 15;
    const int row   = m * 16 + lr;
    const float* dlr = dlow + (size_t)row * R_DIM;

    v8f acc = {0.f,0.f,0.f,0.f,0.f,0.f,0.f,0.f};
    int kb = kw * 40;
    #pragma unroll
    for (int it = 0; it < 10; ++it, kb += 4) {
        v2f a, b;
        a.x = dlr[kb + lhalf];
        a.y = dlr[kb + lhalf + 1];
        b.x = W_dt[(size_t)(kb + lhalf    ) * H_DIM + h0 + lr];
        b.y = W_dt[(size_t)(kb + lhalf + 1) * H_DIM + h0 + lr];
        acc = __builtin_amdgcn_wmma_f32_16x16x4_f32(
                  false, a, false, b, (short)0, acc, false, false);
    }

    __shared__ float lds_part[8 * 256];
    #pragma unroll
    for (int i = 0; i < 8; ++i) {
        const int r = i + ((lane < 16) ? 0 : 8);
        lds_part[wave * 256 + r * 16 + lr] = acc[i];
    }
    __syncthreads();

    // Phase B: 512 (u, hh) pairs, 2 per thread
    #pragma unroll
    for (int j = 0; j < 2; ++j) {
        const int p  = tid + j * 256;
        const int hh = p & 15;
        const int u  = p >> 4;          // 0..31
        const int mu = u >> 4;
        const int r  = u & 15;
        const int hg = h0 + hh;

        const float dtraw = lds_part[(0 + mu) * 256 + r * 16 + hh]
                          + lds_part[(2 + mu) * 256 + r * 16 + hh]
                          + lds_part[(4 + mu) * 256 + r * 16 + hh]
                          + lds_part[(6 + mu) * 256 + r * 16 + hh];

        const float z  = dtraw + b_dt[hg];
        const float dt = (z > 20.0f) ? z : __logf(1.0f + __expf(z));

        const float xv = x[(size_t)u * H_DIM + hg];
        const v4f* hp = (const v4f*)(h + ((size_t)u * H_DIM + hg) * N_DIM);
        const v4f* Ap = (const v4f*)(A_log + (size_t)hg * N_DIM);
        const v4f* Bp = (const v4f*)(Bf + u * N_DIM);
        const v4f* Cp = (const v4f*)(Cf + u * N_DIM);

        float ysum = 0.f;
        #pragma unroll
        for (int q = 0; q < 4; ++q) {
            const v4f hv = __builtin_nontemporal_load(hp + q);  // h read once
            const v4f Al = Ap[q];
            const v4f Bv = Bp[q];
            const v4f Cv = Cp[q];
            float ab;
            ab = __expf(-dt * __expf(Al.x)); ysum += (ab * hv.x + dt * Bv.x * xv) * Cv.x;
            ab = __expf(-dt * __expf(Al.y)); ysum += (ab * hv.y + dt * Bv.y * xv) * Cv.y;
            ab = __expf(-dt * __expf(Al.z)); ysum += (ab * hv.z + dt * Bv.z * xv) * Cv.z;
            ab = __expf(-dt * __expf(Al.w)); ysum += (ab * hv.w + dt * Bv.w * xv) * Cv.w;
        }
        yout[(size_t)u * H_DIM + hg] = ysum + xv * D[hg];
    }
}

// ---------------------------------------------------------------------------
extern "C" void kernel_launch(void* const* d_in, const int* in_sizes, int n_in,
                              void* d_out, int out_size, void* d_ws, size_t ws_size,
                              hipStream_t stream)
{
    const float* x       = (const float*)d_in[0];   // (32,5120)
    const float* W_delta = (const float*)d_in[1];   // (5120,160)
    const float* W_dt    = (const float*)d_in[2];   // (160,5120)
    const float* b_dt    = (const float*)d_in[3];   // (5120,)
    const float* W_B     = (const float*)d_in[4];   // (5120,16)
    const float* W_C     = (const float*)d_in[5];   // (5120,16)
    const float* A_log   = (const float*)d_in[6];   // (5120,16)
    const float* D       = (const float*)d_in[7];   // (5120,)
    const float* h       = (const float*)d_in[8];   // (32,5120,16)
    float*       y       = (float*)d_out;           // (32,5120)

    float* ws   = (float*)d_ws;
    float* part = ws + WS_PART;
    float* dlow = ws + WS_DLOW;
    float* Bf   = ws + WS_BF;
    float* Cf   = ws + WS_CF;

    k1_partial<<<dim3(120), dim3(256), 0, stream>>>(x, W_delta, W_B, W_C, part);
    k1_reduce <<<dim3(24),  dim3(256), 0, stream>>>(part, dlow, Bf, Cf);
    k2_fused  <<<dim3(320), dim3(256), 0, stream>>>(x, W_dt, b_dt, A_log, D, h,
                                                    dlow, Bf, Cf, y);
}